// SpecialSpmm_18167711662236
// MI455X (gfx1250) — compile-verified
//
#include <hip/hip_runtime.h>
#include <hip/hip_bf16.h>
#include <stdint.h>

// COO SpMM: out[row[e]] += values[e] * b[col[e]], E ~ 1.6M, N ~ 100K, d = 128.
// Memory-bound; b (51MB) and out (51MB) are L2-resident on MI455X (192MB L2).
// Strategy: TDM gather-mode async loads pull 8 random b-rows per descriptor
// into LDS (4-deep ring per wave), lanes scale by the edge value and scatter
// with non-returning agent-scope f32 atomics into L2.

typedef __attribute__((ext_vector_type(4))) unsigned int u32x4;
typedef __attribute__((ext_vector_type(8))) int          i32x8;
typedef __attribute__((ext_vector_type(4))) int          i32x4;

#define WAVES_PER_BLOCK 4
#define BLOCK_THREADS   128
#define ROWS_PER_BATCH  8                    // gather mode, 32-bit indices: max 8 rows
#define PIPE_DEPTH      4                    // TDM gathers in flight per wave
#define DIM             128
#define ROW_BYTES       (DIM * 4)                    // 512 B per b-row
#define BUF_BYTES       (ROWS_PER_BATCH * ROW_BYTES) // 4 KB per buffer
#define WAVE_LDS        (PIPE_DEPTH * BUF_BYTES)     // 16 KB per wave

// Issue one TDM gather of up to 8 rows of b[] (32-bit row indices) into LDS.
// Descriptor layout per CDNA5 ISA ch.8 (async_tensor doc, groups 0..3).
__device__ __forceinline__ void tdm_gather_rows(uint32_t lds_off,
                                                const float* __restrict__ bmat,
                                                int N,
                                                const int* __restrict__ colp,
                                                int count) {
    int idx[ROWS_PER_BATCH];
    if (count == ROWS_PER_BATCH) {
#pragma unroll
        for (int i = 0; i < ROWS_PER_BATCH; ++i)
            idx[i] = __builtin_amdgcn_readfirstlane(colp[i]);
    } else {
#pragma unroll
        for (int i = 0; i < ROWS_PER_BATCH; ++i)
            idx[i] = __builtin_amdgcn_readfirstlane(colp[(i < count) ? i : 0]);
    }
    uint64_t ga = (uint64_t)(uintptr_t)bmat;

    u32x4 g0;
    g0.x = 0xC0000001u;                    // count=1 | gather_index_size=32b | gather_mode=1
    g0.y = lds_off;                        // LDS byte address (bits 63:32)
    g0.z = (uint32_t)(ga & 0xFFFFFFFFu);   // global_addr[31:0]
    g0.w = (uint32_t)((ga >> 32) & 0x01FFFFFFu) | 0x80000000u; // addr[56:32] | type=2

    i32x8 g1;
    g1[0] = 0x00020000;                                        // data_size = 4 bytes
    g1[1] = (int)((uint32_t)DIM << 16);                        // tensor_dim0 = 128
    g1[2] = (int)(((uint32_t)N & 0xFFFFu) << 16);              // tensor_dim1 lo16
    g1[3] = (int)(((uint32_t)N >> 16) | ((uint32_t)DIM << 16));// tensor_dim1 hi | tile_dim0
    g1[4] = count & 0xFFFF;                                    // tile_dim1 = #valid indices
    g1[5] = DIM;                                               // tensor_dim0_stride = 128 elems
    g1[6] = 0;
    g1[7] = 0;

    i32x4 g2 = { idx[0], idx[1], idx[2], idx[3] };
    i32x4 g3 = { idx[4], idx[5], idx[6], idx[7] };
    i32x8 gx = { 0, 0, 0, 0, 0, 0, 0, 0 };   // extra operand (6-arg toolchain form)

    __builtin_amdgcn_tensor_load_to_lds(g0, g1, g2, g3, gx, 0);
}

__device__ __forceinline__ void process_edge(const char* cb, int i, int lane,
                                             const int* __restrict__ rowp,
                                             const float* __restrict__ values,
                                             int base, float* __restrict__ out) {
    const int   r = __builtin_amdgcn_readfirstlane(rowp[base + i]);
    const float v = __uint_as_float(
        __builtin_amdgcn_readfirstlane(__float_as_uint(values[base + i])));
    // 32 lanes x float4 cover the 128-wide row: ds_load_b128 each.
    const float4 f = *(const float4*)(cb + i * ROW_BYTES + lane * 16);
    float* o = out + (size_t)r * DIM + lane * 4;
    // Non-returning agent-scope f32 atomics -> L2 RMW units.
    __hip_atomic_fetch_add(o + 0, v * f.x, __ATOMIC_RELAXED, __HIP_MEMORY_SCOPE_AGENT);
    __hip_atomic_fetch_add(o + 1, v * f.y, __ATOMIC_RELAXED, __HIP_MEMORY_SCOPE_AGENT);
    __hip_atomic_fetch_add(o + 2, v * f.z, __ATOMIC_RELAXED, __HIP_MEMORY_SCOPE_AGENT);
    __hip_atomic_fetch_add(o + 3, v * f.w, __ATOMIC_RELAXED, __HIP_MEMORY_SCOPE_AGENT);
}

__global__ void spmm_coo_tdm_kernel(const int* __restrict__ indices,
                                    const float* __restrict__ values,
                                    const float* __restrict__ bmat,
                                    float* __restrict__ out,
                                    int E, int N) {
    extern __shared__ char smem[];
    const int lane = threadIdx.x & 31;
    const int wave = __builtin_amdgcn_readfirstlane((int)(threadIdx.x >> 5));
    const int wave_global = (int)blockIdx.x * WAVES_PER_BLOCK + wave;
    const int total_waves = (int)gridDim.x * WAVES_PER_BLOCK;
    const int num_batches = (E + ROWS_PER_BATCH - 1) / ROWS_PER_BATCH;

    const int* rowp = indices;       // indices[0][:]
    const int* colp = indices + E;   // indices[1][:]

    char* wbuf = smem + (size_t)wave * WAVE_LDS;
    // Flat shared pointers: low 32 bits are the LDS byte offset (ISA aperture rule).
    const uint32_t woff = (uint32_t)(uintptr_t)wbuf;

    if (wave_global >= num_batches) return;   // uniform per wave

    // Prime the pipeline: up to PIPE_DEPTH gathers in flight.
    int issued = 0;
#pragma unroll
    for (int j = 0; j < PIPE_DEPTH; ++j) {
        int pb = wave_global + j * total_waves;
        if (pb < num_batches) {
            int base = pb * ROWS_PER_BATCH;
            int cnt = min(ROWS_PER_BATCH, E - base);
            tdm_gather_rows(woff + j * BUF_BYTES, bmat, N, colp + base, cnt);
            ++issued;
        }
    }

    int slot = 0;
    for (int b = wave_global; b < num_batches; b += total_waves) {
        // Oldest outstanding TDM is this batch; TDM completes in order per wave.
        if (issued >= 4)      __builtin_amdgcn_s_wait_tensorcnt(3);
        else if (issued == 3) __builtin_amdgcn_s_wait_tensorcnt(2);
        else if (issued == 2) __builtin_amdgcn_s_wait_tensorcnt(1);
        else                  __builtin_amdgcn_s_wait_tensorcnt(0);
        asm volatile("" ::: "memory");

        const char* cb = wbuf + slot * BUF_BYTES;
        const int base = b * ROWS_PER_BATCH;
        const int cnt = min(ROWS_PER_BATCH, E - base);
        if (cnt == ROWS_PER_BATCH) {
#pragma unroll
            for (int i = 0; i < ROWS_PER_BATCH; ++i)
                process_edge(cb, i, lane, rowp, values, base, out);
        } else {
            for (int i = 0; i < cnt; ++i)
                process_edge(cb, i, lane, rowp, values, base, out);
        }
        --issued;

        // Refill this slot with the batch PIPE_DEPTH strides ahead.
        int nxt = b + PIPE_DEPTH * total_waves;
        if (nxt < num_batches) {
            int nbase = nxt * ROWS_PER_BATCH;
            int ncnt = min(ROWS_PER_BATCH, E - nbase);
            // TDM is unordered vs DS ops: ensure LDS reads of this buffer drained.
            asm volatile("s_wait_dscnt 0" ::: "memory");
            tdm_gather_rows(woff + slot * BUF_BYTES, bmat, N, colp + nbase, ncnt);
            ++issued;
        }
        slot = (slot + 1) & (PIPE_DEPTH - 1);
    }
}

extern "C" void kernel_launch(void* const* d_in, const int* in_sizes, int n_in,
                              void* d_out, int out_size, void* d_ws, size_t ws_size,
                              hipStream_t stream) {
    const int*   indices = (const int*)d_in[0];   // [2, E] (int32 per harness)
    const float* values  = (const float*)d_in[1]; // [E]
    const float* bmat    = (const float*)d_in[2]; // [N, 128]
    const int E = in_sizes[0] / 2;
    const int N = in_sizes[2] / DIM;
    float* out = (float*)d_out;

    // Accumulation target must start at zero (harness poisons d_out).
    (void)hipMemsetAsync(d_out, 0, (size_t)out_size * sizeof(float), stream);

    const int blocks = 1024;                             // 4096 waves in flight
    const size_t shmem = (size_t)WAVES_PER_BLOCK * WAVE_LDS; // 64 KB / block
    spmm_coo_tdm_kernel<<<blocks, BLOCK_THREADS, shmem, stream>>>(
        indices, values, bmat, out, E, N);
}